// TensorPassingHomogenous_78700980732442
// MI455X (gfx1250) — compile-verified
//
#include <hip/hip_runtime.h>
#include <hip/hip_bf16.h>

typedef float v2f __attribute__((ext_vector_type(2)));
typedef float v8f __attribute__((ext_vector_type(8)));

#define E_TILE   16
#define HID      64
#define RD       384
#define HSTR     68     // h row stride (floats): 4m+k banks -> conflict-free frag loads
#define RSTR     388    // R row stride (floats)
#define NTHREADS 256

// Spherical harmonic constants (match reference)
#define Y0C  0.28209479177387814f   // 0.5/sqrt(pi)
#define C1   0.4886025119029199f    // sqrt(3/(4pi))
#define C2A  1.0925484305920792f    // 0.5*sqrt(15/pi)
#define C2B  0.31539156525252005f   // 0.25*sqrt(5/pi)
#define C2C  0.5462742152960396f    // 0.25*sqrt(15/pi)
// Path normalizations
#define NORM0 0.8862269254527580f   // sqrt(4pi)/sqrt(16)
#define NORM1 1.0854018644190355f   // sqrt(4pi*3/32)
#define INV_SQRT3  0.57735026918962576f
#define INV_SQRT6  0.40824829046386302f
#define INV_SQRT10 0.31622776601683794f

__global__ void zero_f32(float* __restrict__ p, int n) {
  int i = blockIdx.x * blockDim.x + threadIdx.x;
  if (i < n) p[i] = 0.0f;
}

__global__ __launch_bounds__(NTHREADS)
void tp_fused(const float* __restrict__ features,
              const float* __restrict__ dist,
              const float* __restrict__ relv,
              const int*   __restrict__ eidx,
              const float* __restrict__ w1,
              const float* __restrict__ b1,
              const float* __restrict__ w2,
              const float* __restrict__ b2,
              float* __restrict__ out,
              int n_edges)
{
  __shared__ float sh_h[E_TILE * HSTR];   // relu(d*w1+b1), A-matrix of GEMM
  __shared__ float sh_R[E_TILE * RSTR];   // per-edge radial weights (GEMM output)
  __shared__ float sh_Y[E_TILE * 12];     // spherical harmonics l=0..2
  __shared__ float sh_F[E_TILE * 32];     // gathered source features
  __shared__ int   sh_dst[E_TILE];

  const int tid = threadIdx.x;
  const int e0  = blockIdx.x * E_TILE;

  // ---------------- prologue: h, F gather, Y, dst ----------------
  for (int idx = tid; idx < E_TILE * HID; idx += NTHREADS) {
    const int e = idx >> 6, k = idx & (HID - 1);
    const int ee = min(e0 + e, n_edges - 1);
    const float d = dist[ee];
    sh_h[e * HSTR + k] = fmaxf(fmaf(d, w1[k], b1[k]), 0.0f);
  }
  for (int idx = tid; idx < E_TILE * 32; idx += NTHREADS) {
    const int e = idx >> 5, c = idx & 31;
    const int ee = min(e0 + e, n_edges - 1);
    const int s = eidx[ee];                 // edge_index[0][e]
    sh_F[e * 32 + c] = features[s * 32 + c];
  }
  if (tid < E_TILE) {
    const int ee = min(e0 + tid, n_edges - 1);
    float x = relv[ee * 3 + 0], y = relv[ee * 3 + 1], z = relv[ee * 3 + 2];
    const float rn = rsqrtf(x * x + y * y + z * z);
    x *= rn; y *= rn; z *= rn;
    float* Y = &sh_Y[tid * 12];
    Y[0] = Y0C;
    Y[1] = C1 * y;  Y[2] = C1 * z;  Y[3] = C1 * x;     // l=1 order (y,z,x)
    Y[4] = C2A * x * y;
    Y[5] = C2A * y * z;
    Y[6] = C2B * (3.0f * z * z - 1.0f);
    Y[7] = C2A * x * z;
    Y[8] = C2C * (x * x - y * y);
    sh_dst[tid] = eidx[n_edges + ee];       // edge_index[1][e]
  }
  if (tid < HID) __builtin_prefetch(&w2[tid * RD], 0, 1);  // global_prefetch_b8
  __syncthreads();

  // ------- GEMM: R[16x384] = h[16x64] @ w2[64x384] + b2 (fp32 WMMA) -------
  {
    const int wave = tid >> 5;          // 8 waves (wave32)
    const int lane = tid & 31;
    const int m    = lane & 15;         // M row (A) / N col (B,D)
    const int hi   = lane >> 4;         // half-wave select
    const int kk   = hi << 1;           // K sub-offset: lanes16-31 hold K+2,K+3
    const float* hrow = &sh_h[m * HSTR];
#pragma unroll
    for (int t = 0; t < 3; ++t) {       // 24 N-tiles / 8 waves, uniform -> EXEC all 1s
      const int nt  = wave + (t << 3);
      const int col = nt * 16 + m;
      const float bias = b2[col];
      v8f acc = { bias, bias, bias, bias, bias, bias, bias, bias };
      const float* bp = w2 + col;
#pragma unroll
      for (int k0 = 0; k0 < HID; k0 += 4) {
        v2f a, b;
        a.x = hrow[k0 + kk];            // A 16x4: VGPR0=K(+0/+2), VGPR1=K(+1/+3)
        a.y = hrow[k0 + kk + 1];
        b.x = bp[(k0 + kk) * RD];       // B 4x16: same half-wave K split
        b.y = bp[(k0 + kk + 1) * RD];
        acc = __builtin_amdgcn_wmma_f32_16x16x4_f32(
                  false, a, false, b, (short)0, acc, false, false);
      }
#pragma unroll
      for (int r = 0; r < 8; ++r)       // D: VGPR r -> M=r (lo half) / r+8 (hi half)
        sh_R[(r + (hi << 3)) * RSTR + col] = acc[r];
    }
  }
  __syncthreads();

  // ------------- tensor product paths + scatter-add -------------
  {
    const int e = tid >> 4;             // 16 threads per edge
    const int p = tid & 15;
    const bool valid = (e0 + e) < n_edges;
    const float* Rn = &sh_R[e * RSTR];
    const float* F  = &sh_F[e * 32];
    const float* Y  = &sh_Y[e * 12];
    const int dst   = sh_dst[e];
    const float yy = Y[1], yz = Y[2], yx = Y[3];           // l=1 comps (y,z,x)
    const float A2 = Y[4], B2 = Y[5], CZ = Y[6], D2 = Y[7], E2 = Y[8];
#pragma unroll
    for (int cc = 0; cc < 2; ++cc) {
      const int c = p * 2 + cc;         // output component 0..31
      float msg;
      if (c < 8) {
        // scalar outputs: paths (l=0 <- 0x0) and (l=0 <- 1x1 via lf=1, C=delta/sqrt3)
        const int u = c;
        float s0 = 0.0f, s1 = 0.0f;
#pragma unroll
        for (int v = 0; v < 8; ++v) {
          s0 = fmaf(Rn[u * 8 + v], F[v], s0);
          const float* w = &F[8 + 3 * v];
          const float dy = yy * w[0] + yz * w[1] + yx * w[2];
          s1 = fmaf(Rn[64 + u * 8 + v], dy, s1);
        }
        msg = NORM0 * (Y0C * s0 + INV_SQRT3 * s1);
      } else {
        // vector outputs: paths lf=1(from l=0), lf=0, lf=1(cross/sqrt6), lf=2(/sqrt10)
        const int u = (c - 8) / 3;
        const int o = (c - 8) % 3;
        const float kz = INV_SQRT3 * CZ;
        float yo, cx0, cx1, cx2, g0, g1, g2;
        if (o == 0)      { yo = yy; cx0 = 0.0f; cx1 = -yx;      cx2 =  yz;
                           g0 = -kz - E2;      g1 = B2;         g2 = A2; }
        else if (o == 1) { yo = yz; cx0 = yx;   cx1 = 0.0f;     cx2 = -yy;
                           g0 = B2;            g1 = 2.0f * kz;  g2 = D2; }
        else             { yo = yx; cx0 = -yz;  cx1 = yy;       cx2 = 0.0f;
                           g0 = A2;            g1 = D2;         g2 = E2 - kz; }
        float s10 = 0.0f, s110 = 0.0f, s111 = 0.0f, s112 = 0.0f;
#pragma unroll
        for (int v = 0; v < 8; ++v) {
          const float* w  = &F[8 + 3 * v];
          s10 = fmaf(Rn[128 + u * 8 + v], F[v], s10);
          const float* r3 = &Rn[192 + (u * 8 + v) * 3];
          s110 = fmaf(r3[0], w[o], s110);
          const float cr = cx0 * w[0] + cx1 * w[1] + cx2 * w[2];   // (Y1 x F1v)_o
          s111 = fmaf(r3[1], cr, s111);
          const float g  = g0 * w[0] + g1 * w[1] + g2 * w[2];      // l=2 coupling
          s112 = fmaf(r3[2], g, s112);
        }
        msg = NORM1 * (INV_SQRT3 * yo * s10 + (INV_SQRT3 * Y0C) * s110 +
                       INV_SQRT6 * s111 + INV_SQRT10 * s112);
      }
      if (valid)
        __hip_atomic_fetch_add(&out[dst * 32 + c], msg,
                               __ATOMIC_RELAXED, __HIP_MEMORY_SCOPE_AGENT);
    }
  }
}

extern "C" void kernel_launch(void* const* d_in, const int* in_sizes, int n_in,
                              void* d_out, int out_size, void* d_ws, size_t ws_size,
                              hipStream_t stream) {
  (void)n_in; (void)d_ws; (void)ws_size;
  const float* features = (const float*)d_in[0];
  const float* dist     = (const float*)d_in[1];
  const float* relv     = (const float*)d_in[2];
  const int*   eidx     = (const int*)  d_in[3];
  const float* w1       = (const float*)d_in[4];
  const float* b1       = (const float*)d_in[5];
  const float* w2       = (const float*)d_in[6];
  const float* b2       = (const float*)d_in[7];
  float* out = (float*)d_out;
  const int n_edges = in_sizes[1];

  zero_f32<<<(out_size + 255) / 256, 256, 0, stream>>>(out, out_size);
  const int blocks = (n_edges + E_TILE - 1) / E_TILE;
  tp_fused<<<blocks, NTHREADS, 0, stream>>>(features, dist, relv, eidx,
                                            w1, b1, w2, b2, out, n_edges);
}